// AttnDecoderLSTM_64690797412751
// MI455X (gfx1250) — compile-verified
//
#include <hip/hip_runtime.h>
#include <hip/hip_bf16.h>

typedef __attribute__((ext_vector_type(2))) float v2f;
typedef __attribute__((ext_vector_type(8))) float v8f;

#define HS 512
#define BS 64
#define SQ 256
#define VOC 32000

__device__ __forceinline__ float sigmoidf_(float x) { return 1.0f / (1.0f + __expf(-x)); }

__device__ __forceinline__ float wave_sum(float v) {
#pragma unroll
    for (int o = 16; o; o >>= 1) v += __shfl_xor(v, o, 32);
    return v;
}
__device__ __forceinline__ float wave_max(float v) {
#pragma unroll
    for (int o = 16; o; o >>= 1) v = fmaxf(v, __shfl_xor(v, o, 32));
    return v;
}

// hwh[b] = dot(h_top[b], attn_W[0:512]) + attn_b ; one wave per b (64 waves)
__global__ void k_hwh(const float* __restrict__ h_top, const float* __restrict__ attn_W,
                      const float* __restrict__ attn_b, float* __restrict__ hwh) {
    int wave = blockIdx.x * (blockDim.x >> 5) + (threadIdx.x >> 5);
    int lane = threadIdx.x & 31;
    const float* hp = h_top + (size_t)wave * HS;
    float s = 0.f;
    for (int d = lane; d < HS; d += 32) s += hp[d] * attn_W[d];
    s = wave_sum(s);
    if (lane == 0) hwh[wave] = s + attn_b[0];
}

// scores[b*SQ+s] = hwh[b] + dot(enc[s,b,:], We) ; one wave per (s,b) pair
__global__ void k_scores(const float* __restrict__ enc, const float* __restrict__ We,
                         const float* __restrict__ hwh, float* __restrict__ scores) {
    int wave = blockIdx.x * 8 + (threadIdx.x >> 5);
    int lane = threadIdx.x & 31;
    int s = wave >> 6;
    int b = wave & 63;
    const float* e = enc + (size_t)(s * BS + b) * (2 * HS);
    float acc = 0.f;
    for (int d = lane; d < 2 * HS; d += 32) acc += e[d] * We[d];
    acc = wave_sum(acc);
    if (lane == 0) scores[b * SQ + s] = acc + hwh[b];
}

// softmax over each contiguous row of 64 (the reshape(S,-1) is a flat reinterpret)
__global__ void k_softmax(const float* __restrict__ scores, float* __restrict__ norm) {
    int row = blockIdx.x * 8 + (threadIdx.x >> 5);
    int lane = threadIdx.x & 31;
    float e0 = scores[row * 64 + lane];
    float e1 = scores[row * 64 + 32 + lane];
    float m = wave_max(fmaxf(e0, e1));
    float x0 = __expf(e0 - m);
    float x1 = __expf(e1 - m);
    float inv = 1.0f / wave_sum(x0 + x1);
    norm[row * 64 + lane] = x0 * inv;
    norm[row * 64 + 32 + lane] = x1 * inv;
}

// x[b,0:1024] = sum_s norm[s,b]*enc[s,b,:] ; x[b,1024:1536] = input[b,:]
__global__ void k_buildx(const float* __restrict__ norm, const float* __restrict__ enc,
                         const float* __restrict__ inp, float* __restrict__ x) {
    int tid = blockIdx.x * blockDim.x + threadIdx.x;  // 64*1536 threads
    int b = tid / 1536, d = tid % 1536;
    float v;
    if (d < 2 * HS) {
        float acc = 0.f;
        for (int s = 0; s < SQ; ++s)
            acc += norm[s * BS + b] * enc[(size_t)(s * BS + b) * (2 * HS) + d];
        v = acc;
    } else {
        v = inp[b * HS + (d - 2 * HS)];
    }
    x[tid] = v;
}

// C(64xN) = A1(64xK1) @ W1(NxK1)^T [+ A2(64xK2) @ W2(NxK2)^T] + bias1 [+ bias2]
// M-blocked: one wave owns a full 64x16 column block of C (4 accumulators).
// Per k-step: 1 B-fragment load (streamed weight matrix read exactly once from HBM)
// + 4 A-fragment loads (A is tiny, L0/L2-resident) + 4 f32 WMMAs.
__global__ void k_gemm_wmma(const float* __restrict__ A1, const float* __restrict__ W1, int K1,
                            const float* __restrict__ A2, const float* __restrict__ W2, int K2,
                            const float* __restrict__ bias1, const float* __restrict__ bias2,
                            float* __restrict__ C, int N) {
    int nt = blockIdx.x * (blockDim.x >> 5) + (threadIdx.x >> 5);  // N tile index
    int lane = threadIdx.x & 31;
    int r = lane & 15;
    int kh = (lane >> 4) << 1;  // lanes 0-15: K pair {0,1}; lanes 16-31: {2,3}

    v8f acc0 = {}, acc1 = {}, acc2 = {}, acc3 = {};

    {
        const float* a = A1 + (size_t)r * K1 + kh;
        const float* w = W1 + (size_t)(nt * 16 + r) * K1 + kh;
#pragma unroll 2
        for (int k = 0; k < K1; k += 4) {
            v2f bw = *reinterpret_cast<const v2f*>(w + k);
            v2f a0 = *reinterpret_cast<const v2f*>(a + k);
            v2f a1 = *reinterpret_cast<const v2f*>(a + 16 * K1 + k);
            v2f a2 = *reinterpret_cast<const v2f*>(a + 32 * K1 + k);
            v2f a3 = *reinterpret_cast<const v2f*>(a + 48 * K1 + k);
            acc0 = __builtin_amdgcn_wmma_f32_16x16x4_f32(false, a0, false, bw, (short)0, acc0, false, false);
            acc1 = __builtin_amdgcn_wmma_f32_16x16x4_f32(false, a1, false, bw, (short)0, acc1, false, false);
            acc2 = __builtin_amdgcn_wmma_f32_16x16x4_f32(false, a2, false, bw, (short)0, acc2, false, false);
            acc3 = __builtin_amdgcn_wmma_f32_16x16x4_f32(false, a3, false, bw, (short)0, acc3, false, false);
        }
    }
    if (A2) {
        const float* a = A2 + (size_t)r * K2 + kh;
        const float* w = W2 + (size_t)(nt * 16 + r) * K2 + kh;
#pragma unroll 2
        for (int k = 0; k < K2; k += 4) {
            v2f bw = *reinterpret_cast<const v2f*>(w + k);
            v2f a0 = *reinterpret_cast<const v2f*>(a + k);
            v2f a1 = *reinterpret_cast<const v2f*>(a + 16 * K2 + k);
            v2f a2 = *reinterpret_cast<const v2f*>(a + 32 * K2 + k);
            v2f a3 = *reinterpret_cast<const v2f*>(a + 48 * K2 + k);
            acc0 = __builtin_amdgcn_wmma_f32_16x16x4_f32(false, a0, false, bw, (short)0, acc0, false, false);
            acc1 = __builtin_amdgcn_wmma_f32_16x16x4_f32(false, a1, false, bw, (short)0, acc1, false, false);
            acc2 = __builtin_amdgcn_wmma_f32_16x16x4_f32(false, a2, false, bw, (short)0, acc2, false, false);
            acc3 = __builtin_amdgcn_wmma_f32_16x16x4_f32(false, a3, false, bw, (short)0, acc3, false, false);
        }
    }

    int n = nt * 16 + r;
    float bias = bias1 ? bias1[n] : 0.f;
    if (bias2) bias += bias2[n];
    int mrow = (lane >> 4) ? 8 : 0;  // C layout: vgpr v -> M=v (lanes 0-15) / v+8 (lanes 16-31)
#pragma unroll
    for (int v = 0; v < 8; ++v) {
        C[(size_t)(mrow + v) * N + n]      = acc0[v] + bias;
        C[(size_t)(16 + mrow + v) * N + n] = acc1[v] + bias;
        C[(size_t)(32 + mrow + v) * N + n] = acc2[v] + bias;
        C[(size_t)(48 + mrow + v) * N + n] = acc3[v] + bias;
    }
}

// LSTM activations: gates (64x2048, i|f|g|o) -> h, c. Optionally duplicate h to 2nd slot.
__global__ void k_lstm_elem(const float* __restrict__ gates, const float* __restrict__ c_prev,
                            float* __restrict__ h_out, float* __restrict__ h_out2,
                            float* __restrict__ c_out) {
    int idx = blockIdx.x * blockDim.x + threadIdx.x;  // 64*512 threads
    int b = idx >> 9, hh = idx & 511;
    const float* g = gates + (size_t)b * (4 * HS);
    float ig = sigmoidf_(g[hh]);
    float fg = sigmoidf_(g[HS + hh]);
    float gg = tanhf(g[2 * HS + hh]);
    float og = sigmoidf_(g[3 * HS + hh]);
    float c = fg * c_prev[idx] + ig * gg;
    float h = og * tanhf(c);
    c_out[idx] = c;
    h_out[idx] = h;
    if (h_out2) h_out2[idx] = h;
}

extern "C" void kernel_launch(void* const* d_in, const int* in_sizes, int n_in,
                              void* d_out, int out_size, void* d_ws, size_t ws_size,
                              hipStream_t stream) {
    const float* input  = (const float*)d_in[0];
    const float* h0     = (const float*)d_in[1];
    const float* c0     = (const float*)d_in[2];
    const float* enc    = (const float*)d_in[3];
    const float* attn_W = (const float*)d_in[4];
    const float* attn_b = (const float*)d_in[5];
    const float* w_ih0  = (const float*)d_in[6];
    const float* w_hh0  = (const float*)d_in[7];
    const float* b_ih0  = (const float*)d_in[8];
    const float* b_hh0  = (const float*)d_in[9];
    const float* w_ih1  = (const float*)d_in[10];
    const float* w_hh1  = (const float*)d_in[11];
    const float* b_ih1  = (const float*)d_in[12];
    const float* b_hh1  = (const float*)d_in[13];
    const float* out_W  = (const float*)d_in[14];
    const float* out_b  = (const float*)d_in[15];

    float* out        = (float*)d_out;
    float* out_output = out;            // (1,64,512)  32768
    float* out_h      = out + 32768;    // (2,64,512)  65536
    float* out_c      = out + 98304;    // (2,64,512)  65536
    float* out_pred   = out + 163840;   // (64,32000)  2048000

    float* ws     = (float*)d_ws;
    float* hwh    = ws;                  // 64
    float* scores = ws + 64;             // 64*256
    float* norm   = scores + 16384;      // 256*64
    float* xbuf   = norm + 16384;        // 64*1536
    float* gates  = xbuf + 98304;        // 64*2048

    // attention
    k_hwh<<<8, 256, 0, stream>>>(h0, attn_W, attn_b, hwh);
    k_scores<<<2048, 256, 0, stream>>>(enc, attn_W + HS, hwh, scores);
    k_softmax<<<32, 256, 0, stream>>>(scores, norm);
    k_buildx<<<384, 256, 0, stream>>>(norm, enc, input, xbuf);

    // LSTM layer 0: gates = x@w_ih0^T + h0[0]@w_hh0^T + biases   (N=2048 -> 128 waves)
    k_gemm_wmma<<<16, 256, 0, stream>>>(xbuf, w_ih0, 3 * HS, h0, w_hh0, HS,
                                        b_ih0, b_hh0, gates, 4 * HS);
    k_lstm_elem<<<128, 256, 0, stream>>>(gates, c0, out_h, nullptr, out_c);

    // LSTM layer 1: gates = h1@w_ih1^T + h0[1]@w_hh1^T + biases
    k_gemm_wmma<<<16, 256, 0, stream>>>(out_h, w_ih1, HS, h0 + BS * HS, w_hh1, HS,
                                        b_ih1, b_hh1, gates, 4 * HS);
    k_lstm_elem<<<128, 256, 0, stream>>>(gates, c0 + BS * HS, out_h + BS * HS, out_output,
                                         out_c + BS * HS);

    // pred = h2 @ out_W^T + out_b   (dominant GEMM: 64x512x32000 -> 2000 waves)
    k_gemm_wmma<<<250, 256, 0, stream>>>(out_output, out_W, HS, nullptr, nullptr, 0,
                                         out_b, nullptr, out_pred, VOC);
}